// GGNN_8693013807130
// MI455X (gfx1250) — compile-verified
//
#include <hip/hip_runtime.h>
#include <hip/hip_bf16.h>
#include <math.h>

// GGNN: B=8, N=1024, D=64, E=4, T=5, AD=32
#define BB 8
#define NN 1024
#define DD 64
#define EE 4
#define TT 5
#define ADIM 32
#define KTOT (NN * EE)          // 4096
#define AROW (2 * KTOT)         // 8192 (A row stride)
#define KC 64                   // K-chunk for LDS staging
#define LSTRIDE 68              // padded LDS row stride (floats), conflict-free frag reads

typedef __attribute__((ext_vector_type(2))) float v2f;
typedef __attribute__((ext_vector_type(8))) float v8f;

static __device__ __forceinline__ v8f vzero8() {
    v8f v;
#pragma unroll
    for (int i = 0; i < 8; ++i) v[i] = 0.0f;
    return v;
}

static __device__ __forceinline__ v8f wmma_f32(v2f a, v2f b, v8f c) {
    // D = A(16x4,f32) * B(4x16,f32) + C(16x16,f32)
    return __builtin_amdgcn_wmma_f32_16x16x4_f32(false, a, false, b, (short)0, c,
                                                 false, false);
}

// ---------------------------------------------------------------------------
// k1: S[b, e*N+n, o] = sum_i h[b,n,i] * W[e,o,i] + bias[e,o]   (both dirs)
// grid: (32, 2, B*E), block: 256 (8 waves). Each wave: one 16x16 (n,o) tile.
// ---------------------------------------------------------------------------
__global__ __launch_bounds__(256) void ggnn_k1_transform(
    const float* __restrict__ h,      // (B,N,D)
    const float* __restrict__ W_in,   // (E,D,D)
    const float* __restrict__ b_in,   // (E,D)
    const float* __restrict__ W_out,  // (E,D,D)
    const float* __restrict__ b_out,  // (E,D)
    float* __restrict__ S_in,         // (B, E*N, D)
    float* __restrict__ S_out)        // (B, E*N, D)
{
    const int lane = threadIdx.x & 31;
    const int wave = threadIdx.x >> 5;
    const int l    = lane & 15;
    const int half = lane >> 4;

    const int tile = blockIdx.x * 8 + wave;  // 0..255
    const int n0   = (tile >> 2) << 4;       // 0..1008
    const int o0   = (tile & 3) << 4;        // 0..48
    const int dir  = blockIdx.y;
    const int be   = blockIdx.z;
    const int b    = be >> 2;
    const int e    = be & 3;

    const float* W    = (dir ? W_out : W_in) + (size_t)e * DD * DD;
    const float* bias = (dir ? b_out : b_in) + (size_t)e * DD;
    const float* hb   = h + (size_t)b * NN * DD;
    float* S          = (dir ? S_out : S_in) + ((size_t)b * EE + e) * (size_t)NN * DD;

    v8f acc = vzero8();
#pragma unroll
    for (int k = 0; k < DD; k += 4) {
        const int kk = k + 2 * half;
        // A frag: h[n0+l][kk .. kk+1]
        v2f a = *(const v2f*)(hb + (size_t)(n0 + l) * DD + kk);
        // B frag: B[k][o] = W[o][k]  -> W[o0+l][kk .. kk+1]
        v2f bf = *(const v2f*)(W + (size_t)(o0 + l) * DD + kk);
        acc = wmma_f32(a, bf, acc);
    }

    const float bo = bias[o0 + l];
#pragma unroll
    for (int v = 0; v < 8; ++v) {
        const int n = n0 + v + 8 * half;
        S[(size_t)n * DD + o0 + l] = acc[v] + bo;
    }
}

// ---------------------------------------------------------------------------
// k2: a_dir[b,n,d] = sum_m A[b,n, dir*4096 + m] * S_dir[b,m,d]
// grid: (16, 2, B), block: 128 (4 waves). Block computes 64 rows x 64 cols.
// Software-pipelined: next K-chunk streamed into registers while WMMAs run
// from LDS; A frags + transposed-S frags are contiguous, conflict-free b64.
// ---------------------------------------------------------------------------
__global__ __launch_bounds__(128) void ggnn_k2_agg(
    const float* __restrict__ A,     // (B, N, 2*KTOT)
    const float* __restrict__ S_in,  // (B, KTOT, D)
    const float* __restrict__ S_out, // (B, KTOT, D)
    float* __restrict__ a_in,        // (B, N, D)
    float* __restrict__ a_out)       // (B, N, D)
{
    const int mblk = blockIdx.x;  // 64-row block
    const int dir  = blockIdx.y;
    const int b    = blockIdx.z;

    const float* Abase = A + (size_t)b * NN * AROW + (size_t)dir * KTOT;
    const float* Sb    = (dir ? S_out : S_in) + (size_t)b * KTOT * DD;
    float* Cb          = (dir ? a_out : a_in) + (size_t)b * NN * DD;

    __shared__ float As[64 * LSTRIDE];  // As[r][c]   = A[mblk*64+r][k0+c]
    __shared__ float Ts[64 * LSTRIDE];  // Ts[c][kk]  = S[k0+kk][c]

    const int tid  = threadIdx.x;
    const int lane = tid & 31;
    const int wave = tid >> 5;
    const int l    = lane & 15;
    const int half = lane >> 4;
    const int arow = wave * 16 + l;

    // Per-thread staging slice: element index i = tid + j*128, r = i>>4 (row for
    // A, k-row for S), c4 = (i&15)*4 (column group).
    const int r0 = tid >> 4;          // r for j: r = r0 + j*8
    const int c4 = (tid & 15) << 2;

    v8f acc0 = vzero8(), acc1 = vzero8(), acc2 = vzero8(), acc3 = vzero8();

    float4 rA[8], rS[8];
    // Preload chunk 0 into registers.
#pragma unroll
    for (int j = 0; j < 8; ++j) {
        const int r = r0 + j * 8;
        rA[j] = *(const float4*)(Abase + (size_t)(mblk * 64 + r) * AROW + c4);
        rS[j] = *(const float4*)(Sb + (size_t)r * DD + c4);
    }

    for (int k0 = 0; k0 < KTOT; k0 += KC) {
        __syncthreads();  // previous chunk's compute done; LDS free
        // Commit staged registers to LDS (A row-major, S transposed).
#pragma unroll
        for (int j = 0; j < 8; ++j) {
            const int r = r0 + j * 8;
            float* dst = &As[r * LSTRIDE + c4];
            dst[0] = rA[j].x; dst[1] = rA[j].y; dst[2] = rA[j].z; dst[3] = rA[j].w;
            Ts[(c4 + 0) * LSTRIDE + r] = rS[j].x;
            Ts[(c4 + 1) * LSTRIDE + r] = rS[j].y;
            Ts[(c4 + 2) * LSTRIDE + r] = rS[j].z;
            Ts[(c4 + 3) * LSTRIDE + r] = rS[j].w;
        }
        __syncthreads();

        // Kick off the next chunk's global stream (overlaps with WMMAs below).
        const int kn = k0 + KC;
        if (kn < KTOT) {
#pragma unroll
            for (int j = 0; j < 8; ++j) {
                const int r = r0 + j * 8;
                rA[j] = *(const float4*)(Abase + (size_t)(mblk * 64 + r) * AROW + kn + c4);
                rS[j] = *(const float4*)(Sb + (size_t)(kn + r) * DD + c4);
            }
            // Pull A (the streamed 256MB tensor) two chunks ahead into cache.
            if (kn + KC < KTOT) {
                __builtin_prefetch(Abase + (size_t)(mblk * 64 + r0) * AROW + kn + KC + c4, 0, 1);
                __builtin_prefetch(Abase + (size_t)(mblk * 64 + r0 + 32) * AROW + kn + KC + c4, 0, 1);
            }
        }

        // Compute: 16 K-steps of fp32 WMMA from LDS.
#pragma unroll 4
        for (int kk = 0; kk < KC; kk += 4) {
            const int kf = kk + 2 * half;
            v2f a  = *(const v2f*)(&As[arow * LSTRIDE + kf]);
            v2f b0 = *(const v2f*)(&Ts[(0 * 16 + l) * LSTRIDE + kf]);
            v2f b1 = *(const v2f*)(&Ts[(1 * 16 + l) * LSTRIDE + kf]);
            v2f b2 = *(const v2f*)(&Ts[(2 * 16 + l) * LSTRIDE + kf]);
            v2f b3 = *(const v2f*)(&Ts[(3 * 16 + l) * LSTRIDE + kf]);
            acc0 = wmma_f32(a, b0, acc0);
            acc1 = wmma_f32(a, b1, acc1);
            acc2 = wmma_f32(a, b2, acc2);
            acc3 = wmma_f32(a, b3, acc3);
        }
    }

    const int mbase = mblk * 64 + wave * 16 + 8 * half;
#pragma unroll
    for (int v = 0; v < 8; ++v) {
        const size_t m = (size_t)(mbase + v);
        Cb[m * DD +  0 + l] = acc0[v];
        Cb[m * DD + 16 + l] = acc1[v];
        Cb[m * DD + 32 + l] = acc2[v];
        Cb[m * DD + 48 + l] = acc3[v];
    }
}

// ---------------------------------------------------------------------------
// k3: GRU update for one (b,n) row per block of 64 threads.
//   a = [a_in, a_out, h];  r = sig(a Wr^T + br);  z = sig(a Wz^T + bz)
//   hh = tanh([a_in, a_out, r*h] Wh^T + bh);  h' = (1-z)h + z*hh
// ---------------------------------------------------------------------------
__global__ __launch_bounds__(64) void ggnn_k3_gru(
    const float* __restrict__ a_in, const float* __restrict__ a_out,
    const float* __restrict__ h_in,
    const float* __restrict__ Wr, const float* __restrict__ br,
    const float* __restrict__ Wz, const float* __restrict__ bz,
    const float* __restrict__ Wh, const float* __restrict__ bh,
    float* __restrict__ h_out)
{
    const size_t row = blockIdx.x;  // 0 .. B*N-1
    const int o = threadIdx.x;      // 0..63

    __shared__ float acat[3 * DD];
    __shared__ float rh[DD];

    const float hval = h_in[row * DD + o];
    acat[o]           = a_in[row * DD + o];
    acat[DD + o]      = a_out[row * DD + o];
    acat[2 * DD + o]  = hval;
    __syncthreads();

    float accr = br[o], accz = bz[o];
    const float* wr = Wr + (size_t)o * 3 * DD;
    const float* wz = Wz + (size_t)o * 3 * DD;
#pragma unroll 8
    for (int k = 0; k < 3 * DD; ++k) {
        const float av = acat[k];
        accr = fmaf(av, wr[k], accr);
        accz = fmaf(av, wz[k], accz);
    }
    const float r = 1.0f / (1.0f + expf(-accr));
    const float z = 1.0f / (1.0f + expf(-accz));
    rh[o] = r * hval;
    __syncthreads();

    float acch = bh[o];
    const float* wh = Wh + (size_t)o * 3 * DD;
#pragma unroll 8
    for (int k = 0; k < 2 * DD; ++k) acch = fmaf(acat[k], wh[k], acch);
#pragma unroll 8
    for (int k = 0; k < DD; ++k) acch = fmaf(rh[k], wh[2 * DD + k], acch);
    const float hhat = tanhf(acch);

    h_out[row * DD + o] = (1.0f - z) * hval + z * hhat;
}

// ---------------------------------------------------------------------------
// k4: out[b,n] = sum_o tanh(join . Wo1[o,:] + bo1[o]) * Wo2[o]  + bo2
//     join = [h(64), annotation(32)]
// ---------------------------------------------------------------------------
__global__ __launch_bounds__(64) void ggnn_k4_head(
    const float* __restrict__ h,     // (B,N,D)
    const float* __restrict__ anno,  // (B,N,AD)
    const float* __restrict__ Wo1,   // (D, D+AD)
    const float* __restrict__ bo1,   // (D)
    const float* __restrict__ Wo2,   // (1, D)
    const float* __restrict__ bo2,   // (1)
    float* __restrict__ out)         // (B*N)
{
    const size_t row = blockIdx.x;
    const int o = threadIdx.x;

    __shared__ float join[DD + ADIM];
    __shared__ float red[DD];

    join[o] = h[row * DD + o];
    if (o < ADIM) join[DD + o] = anno[row * ADIM + o];
    __syncthreads();

    float acc = bo1[o];
    const float* w = Wo1 + (size_t)o * (DD + ADIM);
#pragma unroll 8
    for (int k = 0; k < DD + ADIM; ++k) acc = fmaf(join[k], w[k], acc);
    red[o] = tanhf(acc) * Wo2[o];
    __syncthreads();

#pragma unroll
    for (int s = 32; s > 0; s >>= 1) {
        if (o < s) red[o] += red[o + s];
        __syncthreads();
    }
    if (o == 0) out[row] = red[0] + bo2[0];
}

// ---------------------------------------------------------------------------
extern "C" void kernel_launch(void* const* d_in, const int* in_sizes, int n_in,
                              void* d_out, int out_size, void* d_ws, size_t ws_size,
                              hipStream_t stream) {
    (void)in_sizes; (void)n_in; (void)out_size; (void)ws_size;

    const float* prop  = (const float*)d_in[0];   // (B,N,D)
    const float* anno  = (const float*)d_in[1];   // (B,N,AD)
    const float* A     = (const float*)d_in[2];   // (B,N,2*N*E)
    const float* W_in  = (const float*)d_in[3];   // (E,D,D)
    const float* b_in  = (const float*)d_in[4];   // (E,D)
    const float* W_out = (const float*)d_in[5];   // (E,D,D)
    const float* b_out = (const float*)d_in[6];   // (E,D)
    const float* Wr    = (const float*)d_in[7];   // (D,3D)
    const float* br    = (const float*)d_in[8];
    const float* Wz    = (const float*)d_in[9];
    const float* bz    = (const float*)d_in[10];
    const float* Wh    = (const float*)d_in[11];
    const float* bh    = (const float*)d_in[12];
    const float* Wo1   = (const float*)d_in[13];  // (D, D+AD)
    const float* bo1   = (const float*)d_in[14];
    const float* Wo2   = (const float*)d_in[15];  // (1,D)
    const float* bo2   = (const float*)d_in[16];

    float* ws = (float*)d_ws;
    const size_t szS = (size_t)BB * EE * NN * DD;  // 2,097,152
    const size_t szH = (size_t)BB * NN * DD;       //   524,288
    float* S_in  = ws;
    float* S_out = S_in + szS;
    float* aIn   = S_out + szS;
    float* aOut  = aIn + szH;
    float* hbuf0 = aOut + szH;
    float* hbuf1 = hbuf0 + szH;

    const float* hcur = prop;
    float* bufs[2] = {hbuf0, hbuf1};

    for (int t = 0; t < TT; ++t) {
        ggnn_k1_transform<<<dim3(32, 2, BB * EE), 256, 0, stream>>>(
            hcur, W_in, b_in, W_out, b_out, S_in, S_out);
        ggnn_k2_agg<<<dim3(NN / 64, 2, BB), 128, 0, stream>>>(
            A, S_in, S_out, aIn, aOut);
        float* hnext = bufs[t & 1];
        ggnn_k3_gru<<<dim3(BB * NN), 64, 0, stream>>>(
            aIn, aOut, hcur, Wr, br, Wz, bz, Wh, bh, hnext);
        hcur = hnext;
    }

    ggnn_k4_head<<<dim3(BB * NN), 64, 0, stream>>>(
        hcur, anno, Wo1, bo1, Wo2, bo2, (float*)d_out);
}